// A2Net_Cond_70308614635790
// MI455X (gfx1250) — compile-verified
//
#include <hip/hip_runtime.h>
#include <cstddef>
#include <cstdint>

typedef float v2f __attribute__((ext_vector_type(2)));
typedef float v8f __attribute__((ext_vector_type(8)));

#define BB 8
#define CC 128
#define TT 16
#define HW 1024
#define SS 16384   // T*H*W
#define MM 64

// -------- fp32 WMMA 16x16x4 : D = A(16x4) * B(4x16) + C(16x16) --------
__device__ __forceinline__ v8f wmma4(v2f a, v2f b, v8f c) {
  return __builtin_amdgcn_wmma_f32_16x16x4_f32(false, a, false, b, (short)0, c,
                                               false, false);
}

__device__ __forceinline__ v8f vzero8() {
  v8f z;
#pragma unroll
  for (int i = 0; i < 8; ++i) z[i] = 0.0f;
  return z;
}

// Dual 64xCC @ CCx16 GEMM over one 16-wide spatial column tile.
// D layout per 16x16 tile: reg r, lanes 0-15 -> M=r, lanes 16-31 -> M=r+8;
// N (spatial) = lane&15.  A operand = weights, B operand = x columns.
// All fragment loads for a k-step are issued as one batch (single clause)
// before the 8 WMMAs so waits drain progressively instead of per-pair.
__device__ __forceinline__ void conv_dual(const float* __restrict__ xs,
                                          const float* __restrict__ Wa,
                                          const float* __restrict__ Wb,
                                          int l15, int hi,
                                          v8f accA[4], v8f accB[4]) {
#pragma unroll 4
  for (int k = 0; k < 32; ++k) {
    const int ch = 4 * k + 2 * hi;            // B layout: K = v + 2*(lane>>4)
    v2f bx;
    bx.x = xs[(size_t)ch * SS];
    bx.y = xs[(size_t)(ch + 1) * SS];
    v2f aa[4], ab[4];
#pragma unroll
    for (int mt = 0; mt < 4; ++mt) {
      const int ro = (mt * 16 + l15) * CC + ch;  // A layout: M = lane&15
      aa[mt] = *(const v2f*)(Wa + ro);
      ab[mt] = *(const v2f*)(Wb + ro);
    }
#pragma unroll
    for (int mt = 0; mt < 4; ++mt) {
      accA[mt] = wmma4(aa[mt], bx, accA[mt]);
      accB[mt] = wmma4(ab[mt], bx, accB[mt]);
    }
  }
}

// -------- kernel 0: zero the accumulation workspace --------
__global__ void k_zero(float* __restrict__ p, int n) {
  int i = blockIdx.x * 256 + threadIdx.x;
  if (i < n) p[i] = 0.0f;
}

// -------- kernel 1: conditioning MLP + softmax -> Cm (8,1024) --------
__global__ __launch_bounds__(256) void k_cond(const float* __restrict__ cond,
                                              const float* __restrict__ W1,
                                              const float* __restrict__ b1,
                                              const float* __restrict__ W2,
                                              const float* __restrict__ b2,
                                              float* __restrict__ CmO) {
  __shared__ float h1s[64];
  __shared__ float red[256];
  const int b = blockIdx.x;
  const int t = threadIdx.x;
  if (t < 64) {
    float s = b1[t];
    const float* cr = cond + b * 64;
    const float* wr = W1 + t * 64;
#pragma unroll 8
    for (int k = 0; k < 64; ++k) s += cr[k] * wr[k];
    h1s[t] = fmaxf(s, 0.0f);
  }
  __syncthreads();
  float v[4];
#pragma unroll
  for (int i = 0; i < 4; ++i) {
    const int j = t + 256 * i;
    float s = b2[j];
    const float* wr = W2 + j * 64;
#pragma unroll 8
    for (int k = 0; k < 64; ++k) s += h1s[k] * wr[k];
    v[i] = fmaxf(s, 0.0f);
  }
  float mx = fmaxf(fmaxf(v[0], v[1]), fmaxf(v[2], v[3]));
  red[t] = mx;
  __syncthreads();
  for (int s = 128; s > 0; s >>= 1) {
    if (t < s) red[t] = fmaxf(red[t], red[t + s]);
    __syncthreads();
  }
  mx = red[0];
  __syncthreads();
  float ls = 0.0f;
#pragma unroll
  for (int i = 0; i < 4; ++i) {
    v[i] = __expf(v[i] - mx);
    ls += v[i];
  }
  red[t] = ls;
  __syncthreads();
  for (int s = 128; s > 0; s >>= 1) {
    if (t < s) red[t] += red[t + s];
    __syncthreads();
  }
  const float inv = 1.0f / red[0];
#pragma unroll
  for (int i = 0; i < 4; ++i) CmO[b * HW + t + 256 * i] = v[i] * inv;
}

// -------- kernel 2: pass 1 --------
// A = Wphi@x + bphi ; E = exp(Wth@x + bth)
// U[b]  += A_tile @ (E*Cm)_tile^T   (WMMA, K = spatial)
// denom[b,n] += sum_s E
__global__ __launch_bounds__(256) void k_pass1(
    const float* __restrict__ x, const float* __restrict__ Wphi,
    const float* __restrict__ bphi, const float* __restrict__ Wth,
    const float* __restrict__ bth, const float* __restrict__ CmG,
    float* __restrict__ denomG, float* __restrict__ UG) {
  __shared__ float cm_s[HW];
  __shared__ float den_l[MM];
  __shared__ float U_l[MM * MM];
  __shared__ float scr[8][16 * 16 + 16];  // per-wave relayout scratch

  const int b = blockIdx.y;
  const int tb = blockIdx.x;  // fixed t-slice: spatial base tb*1024
  const int tid = threadIdx.x;
  const int lane = tid & 31;
  const int w = tid >> 5;
  const int l15 = lane & 15;
  const int hi = lane >> 4;

  for (int i = tid; i < HW; i += 256) cm_s[i] = CmG[b * HW + i];
  for (int i = tid; i < MM * MM; i += 256) U_l[i] = 0.0f;
  if (tid < MM) den_l[tid] = 0.0f;
  __syncthreads();

  const float* xb = x + (size_t)b * CC * SS + (size_t)tb * HW;
  float* sw = &scr[w][0];

  v8f accU[4][4];
#pragma unroll
  for (int mt = 0; mt < 4; ++mt)
#pragma unroll
    for (int nt = 0; nt < 4; ++nt) accU[mt][nt] = vzero8();
  float dsum[4][8];
#pragma unroll
  for (int nt = 0; nt < 4; ++nt)
#pragma unroll
    for (int r = 0; r < 8; ++r) dsum[nt][r] = 0.0f;

#pragma unroll 1
  for (int it = 0; it < 8; ++it) {
    const int ci = it * 8 + w;  // 16-wide column chunk within the 1024 slice
    const float* xs = xb + ci * 16 + l15;
    if (it < 7) __builtin_prefetch(xb + (it + 1) * 128 + w * 16 + l15, 0, 1);

    v8f accP[4], accT[4];
#pragma unroll
    for (int mt = 0; mt < 4; ++mt) {
      accP[mt] = vzero8();
      accT[mt] = vzero8();
    }
    conv_dual(xs, Wphi, Wth, l15, hi, accP, accT);

    const float cmv = cm_s[ci * 16 + l15];

    // --- relayout A tiles (add bias) into WMMA-A fragments via LDS ---
    v2f afrag[4][4];
#pragma unroll
    for (int mt = 0; mt < 4; ++mt) {
      const float* bp = bphi + mt * 16 + 8 * hi;
#pragma unroll
      for (int r = 0; r < 8; ++r) accP[mt][r] += bp[r];
#pragma unroll
      for (int r = 0; r < 8; ++r) sw[(r + 8 * hi) * 16 + l15] = accP[mt][r];
#pragma unroll
      for (int kk = 0; kk < 4; ++kk)
        afrag[mt][kk] = *(const v2f*)(&sw[l15 * 16 + 4 * kk + 2 * hi]);
    }

    // --- E tiles: exp, denom partials, *Cm, relayout to WMMA-B, U GEMM ---
    v2f bfrag[4];
#pragma unroll
    for (int nt = 0; nt < 4; ++nt) {
      const float* bt = bth + nt * 16 + 8 * hi;
#pragma unroll
      for (int r = 0; r < 8; ++r) {
        float e = __expf(accT[nt][r] + bt[r]);
        dsum[nt][r] += e;
        accT[nt][r] = e * cmv;
      }
#pragma unroll
      for (int r = 0; r < 8; ++r) sw[(r + 8 * hi) * 16 + l15] = accT[nt][r];
#pragma unroll
      for (int kk = 0; kk < 4; ++kk)
        bfrag[kk] = *(const v2f*)(&sw[l15 * 16 + 4 * kk + 2 * hi]);
#pragma unroll
      for (int mt = 0; mt < 4; ++mt)
#pragma unroll
        for (int kk = 0; kk < 4; ++kk)
          accU[mt][nt] = wmma4(afrag[mt][kk], bfrag[kk], accU[mt][nt]);
    }
  }

  // --- denom: lane-reduce within 16-lane halves, then LDS/global atomics ---
#pragma unroll
  for (int nt = 0; nt < 4; ++nt)
#pragma unroll
    for (int r = 0; r < 8; ++r) {
      float v = dsum[nt][r];
      v += __shfl_xor(v, 8, 32);
      v += __shfl_xor(v, 4, 32);
      v += __shfl_xor(v, 2, 32);
      v += __shfl_xor(v, 1, 32);
      if (l15 == 0) atomicAdd(&den_l[nt * 16 + r + 8 * hi], v);
    }
#pragma unroll
  for (int mt = 0; mt < 4; ++mt)
#pragma unroll
    for (int nt = 0; nt < 4; ++nt)
#pragma unroll
      for (int r = 0; r < 8; ++r)
        atomicAdd(&U_l[(mt * 16 + r + 8 * hi) * MM + nt * 16 + l15],
                  accU[mt][nt][r]);
  __syncthreads();
  for (int i = tid; i < MM * MM; i += 256)
    atomicAdd(&UG[(size_t)b * MM * MM + i], U_l[i]);
  if (tid < MM) atomicAdd(&denomG[b * MM + tid], den_l[tid]);
}

// -------- kernel 3: AB_T = U / denom --------
__global__ void k_abt(const float* __restrict__ U,
                      const float* __restrict__ denom,
                      float* __restrict__ ABT) {
  const int i = blockIdx.x * 256 + threadIdx.x;  // 8*64*64 = 32768
  const int b = i >> 12;
  ABT[i] = U[i] / denom[b * 64 + (i & 63)];
}

// -------- kernel 4: pass 2 --------
// Recompute theta (exp) + rho convs; V = softmax_n; Z = ABT @ V; attn = E/denom * V
__global__ __launch_bounds__(256) void k_pass2(
    const float* __restrict__ x, const float* __restrict__ Wth,
    const float* __restrict__ bth, const float* __restrict__ Wrho,
    const float* __restrict__ brho, const float* __restrict__ ABTG,
    const float* __restrict__ denomG, float* __restrict__ Z,
    float* __restrict__ Attn) {
  __shared__ float abt_s[MM * MM];
  __shared__ float den_s[MM];
  __shared__ float scr[8][16 * 16 + 16];

  const int b = blockIdx.y;
  const int tb = blockIdx.x;
  const int tid = threadIdx.x;
  const int lane = tid & 31;
  const int w = tid >> 5;
  const int l15 = lane & 15;
  const int hi = lane >> 4;

  for (int i = tid; i < MM * MM; i += 256)
    abt_s[i] = ABTG[(size_t)b * MM * MM + i];
  if (tid < MM) den_s[tid] = denomG[b * MM + tid];
  __syncthreads();

  float rden[4][8];
#pragma unroll
  for (int nt = 0; nt < 4; ++nt)
#pragma unroll
    for (int r = 0; r < 8; ++r)
      rden[nt][r] = 1.0f / den_s[nt * 16 + r + 8 * hi];

  const float* xb = x + (size_t)b * CC * SS + (size_t)tb * HW;
  float* sw = &scr[w][0];

#pragma unroll 1
  for (int it = 0; it < 8; ++it) {
    const int ci = it * 8 + w;
    const float* xs = xb + ci * 16 + l15;
    if (it < 7) __builtin_prefetch(xb + (it + 1) * 128 + w * 16 + l15, 0, 1);

    v8f accR[4], accT[4];
#pragma unroll
    for (int mt = 0; mt < 4; ++mt) {
      accR[mt] = vzero8();
      accT[mt] = vzero8();
    }
    conv_dual(xs, Wrho, Wth, l15, hi, accR, accT);

#pragma unroll
    for (int nt = 0; nt < 4; ++nt) {
      const float* br = brho + nt * 16 + 8 * hi;
      const float* bt = bth + nt * 16 + 8 * hi;
#pragma unroll
      for (int r = 0; r < 8; ++r) {
        accR[nt][r] += br[r];
        accT[nt][r] = __expf(accT[nt][r] + bt[r]);
      }
    }

    // --- softmax over the 64 channels: 32 in-lane values + partner lane^16 ---
    float mx = accR[0][0];
#pragma unroll
    for (int nt = 0; nt < 4; ++nt)
#pragma unroll
      for (int r = 0; r < 8; ++r) mx = fmaxf(mx, accR[nt][r]);
    mx = fmaxf(mx, __shfl_xor(mx, 16, 32));
    float ssum = 0.0f;
#pragma unroll
    for (int nt = 0; nt < 4; ++nt)
#pragma unroll
      for (int r = 0; r < 8; ++r) {
        float e = __expf(accR[nt][r] - mx);
        accR[nt][r] = e;
        ssum += e;
      }
    ssum += __shfl_xor(ssum, 16, 32);
    const float rs = 1.0f / ssum;

    // --- attn = (E / denom) * V, stored transposed (b,n,h,w,t) ---
    const int hw = ci * 16 + l15;
#pragma unroll
    for (int nt = 0; nt < 4; ++nt)
#pragma unroll
      for (int r = 0; r < 8; ++r) {
        const float vv = accR[nt][r] * rs;
        accR[nt][r] = vv;  // V
        const int n = nt * 16 + r + 8 * hi;
        Attn[(((size_t)b * MM + n) * HW + hw) * TT + tb] =
            accT[nt][r] * rden[nt][r] * vv;
      }

    // --- Z = ABT @ V : K = 64 channels, relayout V per nt-tile via LDS ---
    v8f accZ[4];
#pragma unroll
    for (int mt = 0; mt < 4; ++mt) accZ[mt] = vzero8();
#pragma unroll
    for (int nt = 0; nt < 4; ++nt) {
#pragma unroll
      for (int r = 0; r < 8; ++r) sw[(r + 8 * hi) * 16 + l15] = accR[nt][r];
#pragma unroll
      for (int kk = 0; kk < 4; ++kk) {
        v2f bf;
        bf.x = sw[(4 * kk + 2 * hi) * 16 + l15];
        bf.y = sw[(4 * kk + 2 * hi + 1) * 16 + l15];
#pragma unroll
        for (int mt = 0; mt < 4; ++mt) {
          v2f af = *(const v2f*)(&abt_s[(mt * 16 + l15) * MM + nt * 16 +
                                        4 * kk + 2 * hi]);
          accZ[mt] = wmma4(af, bf, accZ[mt]);
        }
      }
    }
#pragma unroll
    for (int mt = 0; mt < 4; ++mt)
#pragma unroll
      for (int r = 0; r < 8; ++r)
        Z[((size_t)b * MM + mt * 16 + r + 8 * hi) * SS + (size_t)tb * HW +
          ci * 16 + l15] = accZ[mt][r];
  }
}

// -------- host launcher --------
extern "C" void kernel_launch(void* const* d_in, const int* in_sizes, int n_in,
                              void* d_out, int out_size, void* d_ws,
                              size_t ws_size, hipStream_t stream) {
  const float* x = (const float*)d_in[0];
  const float* cond = (const float*)d_in[1];
  const float* Wphi = (const float*)d_in[2];
  const float* bphi = (const float*)d_in[3];
  const float* Wth = (const float*)d_in[4];
  const float* bthv = (const float*)d_in[5];
  const float* Wrho = (const float*)d_in[6];
  const float* brho = (const float*)d_in[7];
  const float* W1 = (const float*)d_in[8];
  const float* b1 = (const float*)d_in[9];
  const float* W2 = (const float*)d_in[10];
  const float* b2 = (const float*)d_in[11];

  float* ws = (float*)d_ws;
  float* CmW = ws;              // 8*1024
  float* denom = ws + 8192;     // 8*64
  float* U = ws + 8704;         // 8*64*64
  float* ABT = ws + 41472;      // 8*64*64

  float* Zout = (float*)d_out;
  float* Attn = Zout + (size_t)BB * MM * SS;

  k_zero<<<dim3(130), 256, 0, stream>>>(denom, 512 + 32768);
  k_cond<<<dim3(BB), 256, 0, stream>>>(cond, W1, b1, W2, b2, CmW);
  k_pass1<<<dim3(TT, BB), 256, 0, stream>>>(x, Wphi, bphi, Wth, bthv, CmW,
                                            denom, U);
  k_abt<<<dim3(128), 256, 0, stream>>>(U, denom, ABT);
  k_pass2<<<dim3(TT, BB), 256, 0, stream>>>(x, Wth, bthv, Wrho, brho, ABT,
                                            denom, Zout, Attn);
}